// TopKPoolingLayer_46531675685218
// MI455X (gfx1250) — compile-verified
//
#include <hip/hip_runtime.h>
#include <hip/hip_bf16.h>
#include <stdint.h>

#define NN    100000     // nodes in
#define KK    50000      // nodes kept
#define CC    512        // channels
#define EE    3200000    // edges
#define MPAD  131072     // next pow2 >= NN (2^17) for bitonic sort
#define LOCAL    2048    // elements per LDS-resident sort chunk (16 KB)
#define LTHREADS 1024    // threads per sort block (one pair each)
#define MAXKEY 0xFFFFFFFFFFFFFFFFull

typedef float v2f __attribute__((ext_vector_type(2)));
typedef float v8f __attribute__((ext_vector_type(8)));

// ---------------------------------------------------------------------------
// Normalize w: wn[c] = w[c] / ||w||  (single block, LDS tree reduction)
// ---------------------------------------------------------------------------
__global__ void tk_norm_w(const float* __restrict__ w, float* __restrict__ wn) {
    __shared__ float red[CC];
    int t = threadIdx.x;
    float v = w[t];
    red[t] = v * v;
    __syncthreads();
    for (int s = CC / 2; s > 0; s >>= 1) {
        if (t < s) red[t] += red[t + s];
        __syncthreads();
    }
    wn[t] = v * rsqrtf(red[0]);
}

// ---------------------------------------------------------------------------
// Scoring via V_WMMA_F32_16X16X4_F32.
// One wave per 16-row tile of x. A = 16x4 fp32 tile of x, B = wn broadcast
// across all 16 columns, so D[m][n] == dot(x[m], wn) for every n.
// A/B fp32 layout (ISA 7.12.2): lanes 0-15 K={0,1}, lanes 16-31 K={2,3}.
// D layout: VGPR r, lane 0 -> row r ; lane 16 -> row 8+r.
// Emits tanh score + 64-bit descending-stable sort key.
// ---------------------------------------------------------------------------
__global__ void tk_score_wmma(const float* __restrict__ x,
                              const float* __restrict__ wn,
                              float* __restrict__ scores,
                              unsigned long long* __restrict__ keys) {
    const int lane    = threadIdx.x;          // 0..31 (wave32)
    const int rowBase = blockIdx.x * 16;      // N % 16 == 0
    const int r16     = lane & 15;
    const int phase   = lane >> 4;            // 0: K=0,1  1: K=2,3
    const float* xrow = x + (size_t)(rowBase + r16) * CC + 2 * phase;
    const float* wp   = wn + 2 * phase;

    v8f acc = {};
    #pragma unroll 8
    for (int k = 0; k < CC; k += 4) {
        v2f a = *(const v2f*)(xrow + k);      // x[row][k+2p], x[row][k+2p+1]
        v2f b = *(const v2f*)(wp + k);        // wn broadcast over N columns
        acc = __builtin_amdgcn_wmma_f32_16x16x4_f32(
                  false, a, false, b, (short)0, acc, false, false);
    }

    if (lane == 0 || lane == 16) {            // lane 0: rows 0..7, lane16: 8..15
        const int half = phase * 8;
        #pragma unroll
        for (int r = 0; r < 8; ++r) {
            int row = rowBase + half + r;
            float d = acc[r];                             // x[row] . w/||w||
            scores[row] = tanhf(d);
            unsigned u = __float_as_uint(d);
            u ^= (u >> 31) ? 0xFFFFFFFFu : 0x80000000u;   // ascending monotone
            u = ~u;                                       // descending order
            keys[row] = ((unsigned long long)u << 32) | (unsigned)row;
        }
    }
}

// ---------------------------------------------------------------------------
// Bitonic sort, LDS-fused. Ascending u64 => descending score, ties broken by
// smaller original index (matches jax.lax.top_k stability).
// ---------------------------------------------------------------------------

// Phase A: fully sort each 2048-element chunk in LDS (stages k=2..2048),
// materializing +inf padding for [NN, MPAD). Direction uses the global index
// so chunks alternate asc/desc for the following merge levels.
__global__ void tk_bitonic_local_sort(unsigned long long* __restrict__ keys) {
    __shared__ unsigned long long sh[LOCAL];
    const int t    = threadIdx.x;
    const int base = blockIdx.x * LOCAL;
    for (int e = t; e < LOCAL; e += LTHREADS) {
        int gi = base + e;
        sh[e] = (gi < NN) ? keys[gi] : MAXKEY;
    }
    __syncthreads();
    for (int k = 2; k <= LOCAL; k <<= 1) {
        for (int j = k >> 1; j > 0; j >>= 1) {
            int i = ((t & ~(j - 1)) << 1) | (t & (j - 1));
            int l = i + j;
            bool up = (((base + i) & k) == 0);
            unsigned long long a = sh[i];
            unsigned long long b = sh[l];
            if ((a > b) == up) { sh[i] = b; sh[l] = a; }
            __syncthreads();
        }
    }
    for (int e = t; e < LOCAL; e += LTHREADS) keys[base + e] = sh[e];
}

// Global compare-exchange stage, only for spans j >= LOCAL (cross-block pairs).
__global__ void tk_bitonic_step(unsigned long long* __restrict__ keys, int j, int k) {
    int i = blockIdx.x * blockDim.x + threadIdx.x;
    int l = i ^ j;
    if (l > i) {
        unsigned long long a = keys[i];
        unsigned long long b = keys[l];
        bool up = ((i & k) == 0);
        if ((a > b) == up) { keys[i] = b; keys[l] = a; }
    }
}

// Finish one merge level: spans j = LOCAL/2 .. 1 inside an LDS-resident chunk.
__global__ void tk_bitonic_local_merge(unsigned long long* __restrict__ keys, int k) {
    __shared__ unsigned long long sh[LOCAL];
    const int t    = threadIdx.x;
    const int base = blockIdx.x * LOCAL;
    for (int e = t; e < LOCAL; e += LTHREADS) sh[e] = keys[base + e];
    __syncthreads();
    for (int j = LOCAL >> 1; j > 0; j >>= 1) {
        int i = ((t & ~(j - 1)) << 1) | (t & (j - 1));
        int l = i + j;
        bool up = (((base + i) & k) == 0);
        unsigned long long a = sh[i];
        unsigned long long b = sh[l];
        if ((a > b) == up) { sh[i] = b; sh[l] = a; }
        __syncthreads();
    }
    for (int e = t; e < LOCAL; e += LTHREADS) keys[base + e] = sh[e];
}

// ---------------------------------------------------------------------------
// x_out[i][:] = x[perm[i]][:] * tanh_score[perm[i]]   (float4, coalesced)
// ---------------------------------------------------------------------------
__global__ void tk_gather_scale(const float* __restrict__ x,
                                const unsigned long long* __restrict__ keys,
                                const float* __restrict__ scores,
                                float* __restrict__ xout) {
    int orow = blockIdx.x;                       // 0..KK-1
    int t    = threadIdx.x;                      // 0..127 -> float4 each
    int src  = (int)(unsigned)(keys[orow] & 0xFFFFFFFFull);
    float s  = scores[src];
    const float4* xs = (const float4*)(x    + (size_t)src  * CC);
    float4*       xd = (float4*)      (xout + (size_t)orow * CC);
    float4 v = xs[t];
    v.x *= s; v.y *= s; v.z *= s; v.w *= s;
    xd[t] = v;
}

__global__ void tk_map_init(int* __restrict__ map) {
    int i = blockIdx.x * blockDim.x + threadIdx.x;
    if (i < NN) map[i] = -1;
}

__global__ void tk_map_set(const unsigned long long* __restrict__ keys,
                           int* __restrict__ map) {
    int i = blockIdx.x * blockDim.x + threadIdx.x;
    if (i < KK) map[(int)(unsigned)(keys[i] & 0xFFFFFFFFull)] = i;
}

// Relabel edges; drop (mark -1/-1) if either endpoint was pooled away.
__global__ void tk_edge_filter(const int* __restrict__ ei,
                               const int* __restrict__ map,
                               int* __restrict__ eo) {
    int i = blockIdx.x * blockDim.x + threadIdx.x;
    if (i >= EE) return;
    int a = map[ei[i]];
    int b = map[ei[EE + i]];
    bool valid = (a >= 0) & (b >= 0);
    eo[i]      = valid ? a : -1;
    eo[EE + i] = valid ? b : -1;
}

// ---------------------------------------------------------------------------
extern "C" void kernel_launch(void* const* d_in, const int* in_sizes, int n_in,
                              void* d_out, int out_size, void* d_ws, size_t ws_size,
                              hipStream_t stream) {
    const float* x  = (const float*)d_in[0];   // [N, C]
    const int*   ei = (const int*)  d_in[1];   // [2, E]
    const float* w  = (const float*)d_in[2];   // [C]

    char* ws = (char*)d_ws;                    // ~1.9 MB used
    float*              wn     = (float*)(ws + 0);                     //   2 KB
    float*              scores = (float*)(ws + 4096);                  // 400 KB
    unsigned long long* keys   = (unsigned long long*)(ws + 409600);   //   1 MB
    int*                nmap   = (int*)(ws + 1458176);                 // 400 KB

    float* xout = (float*)d_out;                               // [K, C] fp32
    int*   eout = (int*)((float*)d_out + (size_t)KK * CC);     // [2, E] int32 bits

    // 1) normalized attention vector
    tk_norm_w<<<1, CC, 0, stream>>>(w, wn);

    // 2) scores + sort keys via fp32 WMMA (one wave per 16 rows)
    tk_score_wmma<<<NN / 16, 32, 0, stream>>>(x, wn, scores, keys);

    // 3) bitonic sort of (key,index) pairs, LDS-fused (28 launches total).
    //    Phase A also materializes the +inf padding for [NN, MPAD).
    tk_bitonic_local_sort<<<MPAD / LOCAL, LTHREADS, 0, stream>>>(keys);
    for (int k = LOCAL * 2; k <= MPAD; k <<= 1) {
        for (int j = k >> 1; j >= LOCAL; j >>= 1)
            tk_bitonic_step<<<MPAD / 256, 256, 0, stream>>>(keys, j, k);
        tk_bitonic_local_merge<<<MPAD / LOCAL, LTHREADS, 0, stream>>>(keys, k);
    }

    // 4) gather kept rows, gated by tanh score
    tk_gather_scale<<<KK, CC / 4, 0, stream>>>(x, keys, scores, xout);

    // 5) relabel map + edge filtering
    tk_map_init<<<(NN + 255) / 256, 256, 0, stream>>>(nmap);
    tk_map_set<<<(KK + 255) / 256, 256, 0, stream>>>(keys, nmap);
    tk_edge_filter<<<(EE + 255) / 256, 256, 0, stream>>>(ei, nmap, eout);
}